// TriAttention_38070590111917
// MI455X (gfx1250) — compile-verified
//
#include <hip/hip_runtime.h>
#include <hip/hip_bf16.h>
#include <math.h>

#define B_    64
#define VN_   100
#define QN_   64
#define KN_   64
#define VDIM  2048
#define QDIM  768
#define KGDIM 300
#define HDIM  1024
#define RANK  32
#define HD    32
#define GL    2
#define NVQK  (VN_ * QN_ * KN_)   // 409600
#define KTS   (KN_ * 33)          // kt LDS tile stride-33 (TDM-padded), 2112 floats

typedef __attribute__((ext_vector_type(2))) float v2f;
typedef __attribute__((ext_vector_type(8))) float v8f;
typedef __attribute__((ext_vector_type(4))) unsigned int v4u;
typedef __attribute__((ext_vector_type(8))) int v8i;
typedef __attribute__((ext_vector_type(4))) int v4i;

// D = A(16x4,f32) x B(4x16,f32) + C(16x16,f32)   -> v_wmma_f32_16x16x4_f32
__device__ __forceinline__ v8f wmma_f32(v2f a, v2f b, v8f c) {
  return __builtin_amdgcn_wmma_f32_16x16x4_f32(false, a, false, b, (short)0, c,
                                               false, false);
}

// TDM: load a 2D tile rows x cols (f32) from global (row stride in elements)
// into LDS with 1-DWORD pad per 32-DWORD interval (LDS row stride = 33 floats
// when cols == 32). Wave-uniform args; call from one wave only.
// This toolchain exposes the 6-arg tensor_load_to_lds (clang-23 form):
//   (uint32x4 g0, int32x8 g1, int32x4 g2, int32x4 g3, int32x8 gx, i32 cpol)
__device__ __forceinline__ void tdm_load_2d_f32(const float* gsrc, void* lds_dst,
                                                unsigned rows, unsigned cols,
                                                unsigned row_stride_elems,
                                                unsigned tensor_dim0_elems) {
  unsigned long long ga = (unsigned long long)(uintptr_t)gsrc;
  unsigned lds_a = (unsigned)(uintptr_t)lds_dst;   // low 32 bits = LDS byte addr
  v4u g0;
  g0[0] = 1u;                                       // count=1 (valid), user D#
  g0[1] = lds_a;                                    // lds_addr
  g0[2] = (unsigned)(ga & 0xffffffffu);             // global_addr[31:0]
  g0[3] = (unsigned)((ga >> 32) & 0x1ffffffu) | (2u << 30);  // addr[56:32]|type=2
  v8i g1;
  g1[0] = (int)((2u << 16)      // data_size = 4B
              | (1u << 20)      // pad_enable
              | (4u << 22)      // pad_interval: code 4 = 32 DWORDs
              | (0u << 25));    // pad_amount:   code 0 = 1 DWORD
  g1[1] = (int)((tensor_dim0_elems & 0xffffu) << 16);       // tensor_dim0 lo16
  g1[2] = (int)(((tensor_dim0_elems >> 16) & 0xffffu)       // tensor_dim0 hi16
              | ((rows & 0xffffu) << 16));                  // tensor_dim1 lo16
  g1[3] = (int)((cols & 0xffffu) << 16);                    // dim1 hi=0 | tile_dim0
  g1[4] = (int)(rows & 0xffffu);                            // tile_dim1 | tile_dim2=0
  g1[5] = (int)row_stride_elems;                            // tensor_dim0_stride lo32
  g1[6] = 0;                                                // stride hi | dim1_stride
  g1[7] = 0;
  v4i gz; gz[0] = 0; gz[1] = 0; gz[2] = 0; gz[3] = 0;       // 2D: groups 2/3 unused
  v8i gx; gx[0] = 0; gx[1] = 0; gx[2] = 0; gx[3] = 0;
  gx[4] = 0; gx[5] = 0; gx[6] = 0; gx[7] = 0;
  __builtin_amdgcn_tensor_load_to_lds(g0, g1, gz, gz, gx, 0);
}

// ---------------------------------------------------------------- mask[b,v]
__global__ void mask_kernel(const float* __restrict__ v, int* __restrict__ msk) {
  __shared__ float red[256];
  const int row = blockIdx.x;            // b*VN + v
  const float* p = v + (size_t)row * VDIM;
  float s = 0.f;
  for (int i = threadIdx.x; i < VDIM; i += 256) s += fabsf(p[i]);
  red[threadIdx.x] = s;
  __syncthreads();
  for (int off = 128; off > 0; off >>= 1) {
    if (threadIdx.x < off) red[threadIdx.x] += red[threadIdx.x + off];
    __syncthreads();
  }
  if (threadIdx.x == 0) msk[row] = (red[0] == 0.f) ? 1 : 0;
}

// ------------------------------------------- C[M,N] = relu(A[M,K] @ W[K,N] + b)
// block: 256 thr (8 waves), tile 64x64, K-chunk 16 staged in LDS. M%64==0, N%64==0,
// K%4==0 (2048 / 768 / 300). Main loop unguarded b128 staging; one guarded tail.
__launch_bounds__(256)
__global__ void gemm_bias_relu(const float* __restrict__ A, const float* __restrict__ W,
                               const float* __restrict__ bias, float* __restrict__ C,
                               int M, int N, int K) {
  __shared__ float As[64 * 16];
  __shared__ float Ws[16 * 64];
  const int t    = threadIdx.x;
  const int lane = t & 31;
  const int wv   = t >> 5;               // 0..7
  const int half = lane >> 4;
  const int lid  = lane & 15;
  const int m0 = blockIdx.x * 64;
  const int n0 = blockIdx.y * 64;

  const int t0 = wv * 2, t1 = wv * 2 + 1;          // two 16x16 tiles per wave
  const int mt0 = t0 >> 2, nt0 = t0 & 3;
  const int mt1 = t1 >> 2, nt1 = t1 & 3;

  v8f acc0 = {}, acc1 = {};

  auto mma_step = [&]() {
#pragma unroll
    for (int kk = 0; kk < 16; kk += 4) {
      v2f a0, b0, a1, b1;
      a0.x = As[(mt0 * 16 + lid) * 16 + kk + 2 * half];
      a0.y = As[(mt0 * 16 + lid) * 16 + kk + 2 * half + 1];
      b0.x = Ws[(kk + 2 * half) * 64 + nt0 * 16 + lid];
      b0.y = Ws[(kk + 2 * half + 1) * 64 + nt0 * 16 + lid];
      acc0 = wmma_f32(a0, b0, acc0);
      a1.x = As[(mt1 * 16 + lid) * 16 + kk + 2 * half];
      a1.y = As[(mt1 * 16 + lid) * 16 + kk + 2 * half + 1];
      b1.x = Ws[(kk + 2 * half) * 64 + nt1 * 16 + lid];
      b1.y = Ws[(kk + 2 * half + 1) * 64 + nt1 * 16 + lid];
      acc1 = wmma_f32(a1, b1, acc1);
    }
  };

  const int Kfull = K & ~15;
  for (int k0 = 0; k0 < Kfull; k0 += 16) {
    {  // unguarded vectorized staging: 1 float4 per thread per buffer
      int r  = t >> 2, c4 = (t & 3) * 4;           // As 64x16
      *(float4*)&As[r * 16 + c4] =
          *(const float4*)&A[(size_t)(m0 + r) * K + k0 + c4];
      int r2 = t >> 4, c42 = (t & 15) * 4;         // Ws 16x64
      *(float4*)&Ws[r2 * 64 + c42] =
          *(const float4*)&W[(size_t)(k0 + r2) * N + n0 + c42];
    }
    __syncthreads();
    mma_step();
    __syncthreads();
  }
  if (Kfull < K) {                                 // tail (K=300 case)
    for (int e = t; e < 1024; e += 256) {
      int r = e >> 4, c = e & 15;
      As[e] = (Kfull + c < K) ? A[(size_t)(m0 + r) * K + (Kfull + c)] : 0.f;
      int r2 = e >> 6, c2 = e & 63;
      Ws[e] = (Kfull + r2 < K) ? W[(size_t)(Kfull + r2) * N + (n0 + c2)] : 0.f;
    }
    __syncthreads();
    mma_step();
    __syncthreads();
  }

#pragma unroll
  for (int j = 0; j < 8; ++j) {
    int rr = j + 8 * half;
    {
      int gm = m0 + mt0 * 16 + rr, gn = n0 + nt0 * 16 + lid;
      float x = acc0[j] + bias[gn];
      C[(size_t)gm * N + gn] = x > 0.f ? x : 0.f;
    }
    {
      int gm = m0 + mt1 * 16 + rr, gn = n0 + nt1 * 16 + lid;
      float x = acc1[j] + bias[gn];
      C[(size_t)gm * N + gn] = x > 0.f ? x : 0.f;
    }
  }
}

// ----------------------------------------------------- fused trilinear logits
// grid (7 vtiles, 4 qtiles, 64 b), 512 thr = 16 waves, dynamic LDS ~213KB.
// kt tiles arrive via TDM (tensor_load_to_lds), double-buffered across ranks.
__launch_bounds__(512)
__global__ void tri_kernel(const float* __restrict__ vt, const float* __restrict__ qt,
                           const float* __restrict__ kt, const float* __restrict__ T,
                           const int* __restrict__ msk, float* __restrict__ logits) {
  extern __shared__ float sm[];
  float* vt_s = sm;                  // 16 x 32
  float* qt_s = vt_s + 512;          // 16 x 32
  float* kt_s = qt_s + 512;          // 2 x [64(k) x 33(z, TDM-padded)]
  float* U_s  = kt_s + 2 * KTS;      // 16(v) x 2048(y*64+z*2+g)
  float* P_s  = U_s + 16 * 2048;     // 16(v) x 16(q) x 64(z*2+g)

  const int t    = threadIdx.x;
  const int lane = t & 31;
  const int w    = t >> 5;           // 0..15 (= local v in stages P/logits)
  const int half = lane >> 4;
  const int lid  = lane & 15;
  const int v0 = blockIdx.x * 16;
  const int q0 = blockIdx.y * 16;
  const int b  = blockIdx.z;

  v8f zero = {};
  v8f acc[8];
#pragma unroll
  for (int i = 0; i < 8; ++i) acc[i] = zero;

  // prologue: TDM-prefetch rank 0's kt tile [64k x 32z] -> LDS buffer 0
  if (w == 0) {
    tdm_load_2d_f32(kt + (size_t)(b * KN_) * HDIM, kt_s, KN_, HD, HDIM, HDIM);
  }

  for (int r = 0; r < RANK; ++r) {
    const float* Tr = T + (size_t)r * (HD * HD * HD * GL);   // [32x][2048c]
    if (r + 1 < RANK)
      __builtin_prefetch(T + (size_t)(r + 1) * (HD * HD * HD * GL) + t * 32, 0, 0);

    // pipeline: issue next rank's TDM, then wait for current buffer (in-order)
    if (w == 0) {
      if (r + 1 < RANK) {
        tdm_load_2d_f32(kt + (size_t)(b * KN_) * HDIM + (r + 1) * HD,
                        kt_s + ((r + 1) & 1) * KTS, KN_, HD, HDIM, HDIM);
        __builtin_amdgcn_s_wait_tensorcnt(1);
      } else {
        __builtin_amdgcn_s_wait_tensorcnt(0);
      }
    }
    // per-rank vt/qt tiles (one float per thread, coalesced)
    {
      int rv = t >> 5, cx = t & 31;
      int vg = v0 + rv;
      vt_s[t] = (vg < VN_) ? vt[((size_t)(b * VN_ + vg)) * HDIM + r * HD + cx] : 0.f;
      qt_s[t] = qt[((size_t)(b * QN_ + q0 + rv)) * HDIM + r * HD + cx];
    }
    __syncthreads();
    const float* ktc = kt_s + (r & 1) * KTS;       // current [k][33] tile

    // Stage U:  U[v,c] = sum_x vt[v,x] * Tr[x,c] ; wave w owns c in [128w,128w+128)
#pragma unroll
    for (int ci = 0; ci < 8; ++ci) {
      int c0 = w * 128 + ci * 16;
      v8f u = zero;
#pragma unroll
      for (int kk = 0; kk < 32; kk += 4) {
        v2f a, bb;
        a.x  = vt_s[lid * 32 + kk + 2 * half];
        a.y  = vt_s[lid * 32 + kk + 2 * half + 1];
        bb.x = Tr[(size_t)(kk + 2 * half) * 2048 + c0 + lid];
        bb.y = Tr[(size_t)(kk + 2 * half + 1) * 2048 + c0 + lid];
        u = wmma_f32(a, bb, u);
      }
#pragma unroll
      for (int j = 0; j < 8; ++j)
        U_s[(size_t)(j + 8 * half) * 2048 + c0 + lid] = u[j];
    }
    __syncthreads();

    // Stage P (wave w -> v=w):  P[w,q,zg] = sum_y qt[q,y] * U[w, y*64+zg]
#pragma unroll
    for (int ni = 0; ni < 4; ++ni) {
      int n0 = ni * 16;
      v8f p = zero;
#pragma unroll
      for (int kk = 0; kk < 32; kk += 4) {
        v2f a, bb;
        a.x  = qt_s[lid * 32 + kk + 2 * half];
        a.y  = qt_s[lid * 32 + kk + 2 * half + 1];
        bb.x = U_s[w * 2048 + (kk + 2 * half) * 64 + n0 + lid];
        bb.y = U_s[w * 2048 + (kk + 2 * half + 1) * 64 + n0 + lid];
        p = wmma_f32(a, bb, p);
      }
#pragma unroll
      for (int j = 0; j < 8; ++j)
        P_s[w * 1024 + (j + 8 * half) * 64 + n0 + lid] = p[j];
    }
    __syncthreads();

    // Stage logits (wave w -> v=w): acc[nt] += P[w,q,z*2+g] x kt[k,z]
    // B frag: B[z][k] = ktc[k*33 + z]; stride 33 is coprime with 64 banks.
#pragma unroll
    for (int nt = 0; nt < 8; ++nt) {
      int g  = nt & 1;
      int kb = (nt >> 1) * 16;
#pragma unroll
      for (int kk = 0; kk < 32; kk += 4) {
        v2f a, bb;
        a.x  = P_s[w * 1024 + lid * 64 + (kk + 2 * half) * 2 + g];
        a.y  = P_s[w * 1024 + lid * 64 + (kk + 2 * half + 1) * 2 + g];
        bb.x = ktc[(kb + lid) * 33 + kk + 2 * half];
        bb.y = ktc[(kb + lid) * 33 + kk + 2 * half + 1];
        acc[nt] = wmma_f32(a, bb, acc[nt]);
      }
    }
    __syncthreads();
  }

  // write logits tile (with padding mask)
  const int vg = v0 + w;
  if (vg < VN_) {
    const int masked = msk[b * VN_ + vg];
#pragma unroll
    for (int nt = 0; nt < 8; ++nt) {
      int g = nt & 1, kb = (nt >> 1) * 16;
#pragma unroll
      for (int j = 0; j < 8; ++j) {
        int qg = q0 + j + 8 * half;
        int kk = kb + lid;
        float val = masked ? -INFINITY : acc[nt][j];
        logits[((((size_t)b * VN_ + vg) * QN_ + qg) * KN_ + kk) * GL + g] = val;
      }
    }
  }
}

// ------------------------------------------------------------------ softmax
__global__ void smax_max(const float* __restrict__ logits, float* __restrict__ Mx) {
  __shared__ float red[256];
  const int bg = blockIdx.x, b = bg >> 1, g = bg & 1;
  const float* p = logits + (size_t)b * NVQK * GL + g;
  float m = -INFINITY;
  for (int i = threadIdx.x; i < NVQK; i += 256) m = fmaxf(m, p[(size_t)i * GL]);
  red[threadIdx.x] = m;
  __syncthreads();
  for (int off = 128; off > 0; off >>= 1) {
    if (threadIdx.x < off)
      red[threadIdx.x] = fmaxf(red[threadIdx.x], red[threadIdx.x + off]);
    __syncthreads();
  }
  if (threadIdx.x == 0) Mx[bg] = red[0];
}

__global__ void smax_sum(const float* __restrict__ logits, const float* __restrict__ Mx,
                         float* __restrict__ S) {
  __shared__ float red[256];
  const int bg = blockIdx.x, b = bg >> 1, g = bg & 1;
  const float mx = Mx[bg];
  const float* p = logits + (size_t)b * NVQK * GL + g;
  float s = 0.f;
  for (int i = threadIdx.x; i < NVQK; i += 256) s += __expf(p[(size_t)i * GL] - mx);
  red[threadIdx.x] = s;
  __syncthreads();
  for (int off = 128; off > 0; off >>= 1) {
    if (threadIdx.x < off) red[threadIdx.x] += red[threadIdx.x + off];
    __syncthreads();
  }
  if (threadIdx.x == 0) S[bg] = red[0];
}

__global__ void smax_write(const float* __restrict__ logits, const float* __restrict__ Mx,
                           const float* __restrict__ S, float* __restrict__ Pout,
                           size_t total) {
  size_t i = (size_t)blockIdx.x * blockDim.x + threadIdx.x;
  const size_t stride = (size_t)gridDim.x * blockDim.x;
  for (; i < total; i += stride) {
    size_t b = i / ((size_t)NVQK * GL);
    int g = (int)(i & 1);
    int bg = (int)b * GL + g;
    Pout[i] = __expf(logits[i] - Mx[bg]) / S[bg];
  }
}

// -------------------------------------------------------------------- launch
extern "C" void kernel_launch(void* const* d_in, const int* in_sizes, int n_in,
                              void* d_out, int out_size, void* d_ws, size_t ws_size,
                              hipStream_t stream) {
  (void)in_sizes; (void)n_in; (void)out_size; (void)ws_size;
  const float* v   = (const float*)d_in[0];
  const float* q   = (const float*)d_in[1];
  const float* kg  = (const float*)d_in[2];
  const float* Wv  = (const float*)d_in[3];
  const float* bv  = (const float*)d_in[4];
  const float* Wq  = (const float*)d_in[5];
  const float* bq  = (const float*)d_in[6];
  const float* Wkg = (const float*)d_in[7];
  const float* bkg = (const float*)d_in[8];
  const float* T   = (const float*)d_in[9];

  float* out = (float*)d_out;
  const size_t half_out = (size_t)B_ * VN_ * QN_ * KN_ * GL;   // 52,428,800
  float* p_out      = out;
  float* logits_out = out + half_out;

  char* ws = (char*)d_ws;
  float* vt  = (float*)ws; ws += (size_t)B_ * VN_ * HDIM * sizeof(float);
  float* qt  = (float*)ws; ws += (size_t)B_ * QN_ * HDIM * sizeof(float);
  float* kt  = (float*)ws; ws += (size_t)B_ * KN_ * HDIM * sizeof(float);
  int*   msk = (int*)ws;   ws += (size_t)B_ * VN_ * sizeof(int);
  float* Mx  = (float*)ws; ws += (size_t)B_ * GL * sizeof(float);
  float* S   = (float*)ws;

  mask_kernel<<<B_ * VN_, 256, 0, stream>>>(v, msk);

  gemm_bias_relu<<<dim3(B_ * VN_ / 64, HDIM / 64), 256, 0, stream>>>(
      v, Wv, bv, vt, B_ * VN_, HDIM, VDIM);
  gemm_bias_relu<<<dim3(B_ * QN_ / 64, HDIM / 64), 256, 0, stream>>>(
      q, Wq, bq, qt, B_ * QN_, HDIM, QDIM);
  gemm_bias_relu<<<dim3(B_ * KN_ / 64, HDIM / 64), 256, 0, stream>>>(
      kg, Wkg, bkg, kt, B_ * KN_, HDIM, KGDIM);

  const size_t lds_bytes =
      (size_t)(512 + 512 + 2 * KTS + 16 * 2048 + 16 * 16 * 64) * sizeof(float);
  tri_kernel<<<dim3((VN_ + 15) / 16, QN_ / 16, B_), 512, lds_bytes, stream>>>(
      vt, qt, kt, T, msk, logits_out);

  smax_max<<<B_ * GL, 256, 0, stream>>>(logits_out, Mx);
  smax_sum<<<B_ * GL, 256, 0, stream>>>(logits_out, Mx, S);
  smax_write<<<4096, 256, 0, stream>>>(logits_out, Mx, S, p_out, half_out);
}